// OneHotProjEmbedding_21062519619650
// MI455X (gfx1250) — compile-verified
//
#include <hip/hip_runtime.h>
#include <hip/hip_bf16.h>
#include <stdint.h>

#define NUM_LABELS 100000
#define EMBED      128
#define TOKENS     (16 * 4096)

typedef __attribute__((ext_vector_type(4))) float v4f;
typedef __attribute__((ext_vector_type(4))) int   v4i;

#if __has_builtin(__builtin_amdgcn_global_load_async_to_lds_b128)
#define HAVE_ASYNC_LDS 1
typedef __attribute__((address_space(1))) v4i* gptr_v4i;
typedef __attribute__((address_space(3))) v4i* lptr_v4i;
#endif

// ---------------------------------------------------------------------------
// Kernel 1: Wt[n][e] = W[e][n] + bias[e]
// Tile: 32 labels (n) x 128 embed (e) per block, stored UNtransposed in LDS as
// tile[e][n]. Load phase: 4 async global->LDS b128 ops per thread (each wave
// instruction moves 512B, W reads contiguous along n). Transpose happens on the
// LDS *read* side: tile[e][nl] with row stride 33 words -> gcd(33,64)=1 ->
// bank-conflict-free when lanes sweep e. Store phase writes Wt coalesced along
// e (512B contiguous per label row), bias folded in.
// ---------------------------------------------------------------------------
__global__ __launch_bounds__(256) void transpose_bias_kernel(
    const float* __restrict__ W, const float* __restrict__ bias,
    float* __restrict__ Wt) {
  __shared__ float tile[128][33];  // pad 32->33: transposed reads conflict-free

  const int tid = threadIdx.x;
  const int n0  = blockIdx.x * 32;

  // ---- load phase: e = tid>>3 + 32*i (0..127), n-chunk = 4*(tid&7) ----
  const int chunk = (tid & 7) * 4;   // 0,4,...,28
  const int e_ld  = tid >> 3;        // 0..31
#if defined(HAVE_ASYNC_LDS)
  for (int i = 0; i < 4; ++i) {
    const int e = e_ld + i * 32;
    const float* g = W + (size_t)e * NUM_LABELS + (n0 + chunk);
    float* l = &tile[e][chunk];
    // cpol=1 -> TH_NT: W is read exactly once; don't pollute L2 (keep it for Wt).
    __builtin_amdgcn_global_load_async_to_lds_b128((gptr_v4i)g, (lptr_v4i)l, 0, 1);
  }
#if __has_builtin(__builtin_amdgcn_s_wait_asynccnt)
  __builtin_amdgcn_s_wait_asynccnt(0);
#else
  asm volatile("s_wait_asynccnt 0" ::: "memory");
#endif
#else
  for (int i = 0; i < 4; ++i) {
    const int e = e_ld + i * 32;
    const v4f vv = __builtin_nontemporal_load(
        (const v4f*)(W + (size_t)e * NUM_LABELS + (n0 + chunk)));
    tile[e][chunk + 0] = vv.x;
    tile[e][chunk + 1] = vv.y;
    tile[e][chunk + 2] = vv.z;
    tile[e][chunk + 3] = vv.w;
  }
#endif
  __syncthreads();

  // ---- store phase: e = tid&127 (constant per thread), nl = tid>>7 + 2*i ----
  const int e_st    = tid & 127;
  const int nl_base = tid >> 7;     // 0..1
  const float be    = bias[e_st];   // loaded once, reused 16x
  for (int i = 0; i < 16; ++i) {
    const int nl = nl_base + i * 2;
    Wt[(size_t)(n0 + nl) * EMBED + e_st] = tile[e_st][nl] + be;  // transposed LDS read
  }
}

// ---------------------------------------------------------------------------
// Kernel 2: out[t,:] = Wt[idx[t],:]   (bias already folded into Wt)
// One wave32 per token: lane l moves float4 #l -> one 512B fully-coalesced
// global_load_b128 burst per row (L2 hit on Wt), streamed out with NT stores.
// ---------------------------------------------------------------------------
__global__ __launch_bounds__(256) void gather_kernel(
    const int* __restrict__ idx, const float* __restrict__ Wt,
    float* __restrict__ out) {
  const int wave  = threadIdx.x >> 5;
  const int lane  = threadIdx.x & 31;
  const int token = blockIdx.x * 8 + wave;

  const int row = idx[token];
  const v4f* src = (const v4f*)(Wt + (size_t)row * EMBED);
  v4f v = src[lane];
  v4f* dst = (v4f*)(out + (size_t)token * EMBED);
  __builtin_nontemporal_store(v, dst + lane);  // 32MB streaming write: keep L2 for Wt
}

// ---------------------------------------------------------------------------
// Fallback (workspace too small): direct strided gather from W.
// ---------------------------------------------------------------------------
__global__ __launch_bounds__(256) void direct_gather_kernel(
    const int* __restrict__ idx, const float* __restrict__ W,
    const float* __restrict__ bias, float* __restrict__ out) {
  const size_t gid = (size_t)blockIdx.x * blockDim.x + threadIdx.x;
  const int t = (int)(gid >> 7);
  const int e = (int)(gid & 127);
  const int row = idx[t];
  out[gid] = W[(size_t)e * NUM_LABELS + row] + bias[e];
}

extern "C" void kernel_launch(void* const* d_in, const int* in_sizes, int n_in,
                              void* d_out, int out_size, void* d_ws,
                              size_t ws_size, hipStream_t stream) {
  (void)in_sizes; (void)n_in; (void)out_size;
  const int*   X = (const int*)d_in[0];    // [B,S,1] flat = 65536 indices
  const float* W = (const float*)d_in[1];  // [128, 100000]
  const float* b = (const float*)d_in[2];  // [128]
  float* out = (float*)d_out;              // [16,4096,128]

  const size_t need = (size_t)NUM_LABELS * EMBED * sizeof(float);  // 51.2 MB
  if (d_ws != nullptr && ws_size >= need) {
    float* Wt = (float*)d_ws;
    transpose_bias_kernel<<<NUM_LABELS / 32, 256, 0, stream>>>(W, b, Wt);
    gather_kernel<<<TOKENS / 8, 256, 0, stream>>>(X, Wt, out);
  } else {
    direct_gather_kernel<<<(TOKENS * EMBED) / 256, 256, 0, stream>>>(X, W, b, out);
  }
}